// ContinuousStateKernel_83159156785381
// MI455X (gfx1250) — compile-verified
//
#include <hip/hip_runtime.h>
#include <math.h>

// ---------------------------------------------------------------------------
// Problem constants (from reference): B=2, L=512 -> BL=1024, d_inner=1024,
// d_state=64, hidden=256.
// ---------------------------------------------------------------------------
#define BL_TOT   1024
#define DIN      1024
#define NST      64
#define HMID     256

typedef __attribute__((ext_vector_type(2))) float v2f;
typedef __attribute__((ext_vector_type(8))) float v8f;

// ---------------------------------------------------------------------------
// Kernel 0: A = -exp(A_log)   (d_inner x d_state = 64K elements)
// ---------------------------------------------------------------------------
__global__ void neg_exp_kernel(const float* __restrict__ a_log,
                               float* __restrict__ a_out, int n) {
  int i = blockIdx.x * blockDim.x + threadIdx.x;
  if (i < n) a_out[i] = -__expf(a_log[i]);
}

// ---------------------------------------------------------------------------
// Kernel 1: generic WMMA f32 GEMM:  Out[M,N] = A[M,K] * W[N,K]^T  (+epilogue)
//   mode 0: plain            (Bm / Cm projections)
//   mode 1: gelu(acc + bias) (dt hidden layer)
//   mode 2: softplus(acc + bias) * 0.1 (dt output layer)
// One wave (32 lanes) computes one 16x16 output tile with V_WMMA_F32_16X16X4_F32.
// 32-bit A-matrix layout (ISA 7.12.2): lanes 0-15 hold K={k0,k0+1} for M=lane,
// lanes 16-31 hold K={k0+2,k0+3}.  B (4x16) mirrors this with N=lane&15.
// C/D: 8 VGPRs, lane = {half*? } -> row = half*8 + vgpr, col = lane&15.
// ---------------------------------------------------------------------------
__global__ void wmma_gemm_xwt(const float* __restrict__ Amat,
                              const float* __restrict__ W,
                              const float* __restrict__ bias,
                              float* __restrict__ Out,
                              int M, int N, int K, int mode) {
  const int wid  = (blockIdx.x * blockDim.x + threadIdx.x) >> 5;
  const int lane = threadIdx.x & 31;
  const int ntiles = N >> 4;
  const int tm = wid / ntiles;
  const int tn = wid - tm * ntiles;
  if (tm >= (M >> 4)) return;           // uniform per-wave, EXEC stays all-1s

  const int half = lane >> 4;           // 0: K lanes {k,k+1}, 1: {k+2,k+3}
  const int l15  = lane & 15;
  const int arow = tm * 16 + l15;       // A row handled by this lane
  const int bcol = tn * 16 + l15;       // W row (output col) handled by lane
  const int kb   = half * 2;

  v8f acc = {};
  for (int k0 = 0; k0 < K; k0 += 4) {
    v2f a, b;
    a.x = Amat[(size_t)arow * K + k0 + kb];
    a.y = Amat[(size_t)arow * K + k0 + kb + 1];
    b.x = W[(size_t)bcol * K + k0 + kb];
    b.y = W[(size_t)bcol * K + k0 + kb + 1];
    acc = __builtin_amdgcn_wmma_f32_16x16x4_f32(
        /*neg_a=*/false, a, /*neg_b=*/false, b,
        /*c_mod=*/(short)0, acc, /*reuse_a=*/false, /*reuse_b=*/false);
  }

  const int col = tn * 16 + l15;
#pragma unroll
  for (int vg = 0; vg < 8; ++vg) {
    const int row = tm * 16 + half * 8 + vg;
    float v = acc[vg];
    if (mode == 1) {                    // exact GELU
      v += bias[col];
      v = 0.5f * v * (1.0f + erff(v * 0.70710678118654752f));
    } else if (mode == 2) {             // softplus * (DT_BASE*DT_SCALE)
      v += bias[col];
      v = (v > 20.0f) ? v : log1pf(__expf(v));
      v *= 0.1f;
    }
    Out[(size_t)row * N + col] = v;
  }
}

// ---------------------------------------------------------------------------
// Kernel 2: fused state update (memory-bound streaming pass).
// One wave32 per (bl, d) row: n=64 states -> 2 per lane (float2, coalesced
// 64-bit loads/stores on h / h_new).  y-reduction over n done with shfl_xor.
//   h_new = exp(dt*A) * h + (dt*x) * Bm
//   y     = sum_n h_new * Cm + D*x
// ---------------------------------------------------------------------------
__global__ void fused_state_kernel(const float* __restrict__ x,
                                   const float* __restrict__ h,
                                   const float* __restrict__ A,
                                   const float* __restrict__ Bm,
                                   const float* __restrict__ Cm,
                                   const float* __restrict__ Dv,
                                   const float* __restrict__ dt,
                                   float* __restrict__ y,
                                   float* __restrict__ hn) {
  const int wave = (int)((blockIdx.x * blockDim.x + threadIdx.x) >> 5);
  const int lane = threadIdx.x & 31;
  // wave indexes (bl, d) row-major: wave = bl*DIN + d
  const int bl = wave >> 10;
  const int d  = wave & (DIN - 1);

  const float dtv = dt[wave];
  const float xv  = x[wave];
  const float bx  = dtv * xv;

  const float2 Av = ((const float2*)(A  + (size_t)d  * NST))[lane];
  const float2 Bv = ((const float2*)(Bm + (size_t)bl * NST))[lane];
  const float2 Cv = ((const float2*)(Cm + (size_t)bl * NST))[lane];
  const float2 hv = ((const float2*)(h  + (size_t)wave * NST))[lane];

  float2 hnv;
  hnv.x = __expf(dtv * Av.x) * hv.x + bx * Bv.x;
  hnv.y = __expf(dtv * Av.y) * hv.y + bx * Bv.y;
  ((float2*)(hn + (size_t)wave * NST))[lane] = hnv;

  float part = hnv.x * Cv.x + hnv.y * Cv.y;
#pragma unroll
  for (int off = 16; off > 0; off >>= 1)
    part += __shfl_xor(part, off, 32);

  if (lane == 0) y[wave] = part + Dv[d] * xv;
}

// ---------------------------------------------------------------------------
// Host launcher
// ---------------------------------------------------------------------------
extern "C" void kernel_launch(void* const* d_in, const int* in_sizes, int n_in,
                              void* d_out, int out_size, void* d_ws, size_t ws_size,
                              hipStream_t stream) {
  (void)in_sizes; (void)n_in; (void)out_size; (void)ws_size;
  const float* x     = (const float*)d_in[0];  // (2,512,1024)
  const float* h     = (const float*)d_in[1];  // (2,512,1024,64)
  const float* A_log = (const float*)d_in[2];  // (1024,64)
  const float* B_w   = (const float*)d_in[3];  // (64,1024)
  const float* C_w   = (const float*)d_in[4];  // (64,1024)
  const float* Dv    = (const float*)d_in[5];  // (1024,)
  const float* dt_w1 = (const float*)d_in[6];  // (256,1024)
  const float* dt_b1 = (const float*)d_in[7];  // (256,)
  const float* dt_w2 = (const float*)d_in[8];  // (1024,256)
  const float* dt_b2 = (const float*)d_in[9];  // (1024,)

  float* y  = (float*)d_out;                   // (2,512,1024) = 1048576
  float* hn = y + (size_t)BL_TOT * DIN;        // (2,512,1024,64)

  float* ws   = (float*)d_ws;
  float* A    = ws;                                        // 64K
  float* Bm   = A  + (size_t)DIN * NST;                    // 64K  (BL x 64)
  float* Cm   = Bm + (size_t)BL_TOT * NST;                 // 64K  (BL x 64)
  float* G1   = Cm + (size_t)BL_TOT * NST;                 // 256K (BL x 256)
  float* dt   = G1 + (size_t)BL_TOT * HMID;                // 1M   (BL x 1024)

  // A = -exp(A_log)
  {
    int n = DIN * NST;
    neg_exp_kernel<<<(n + 255) / 256, 256, 0, stream>>>(A_log, A, n);
  }

  // Bm = x @ B_w^T   (M=1024, N=64, K=1024)
  {
    int waves = (BL_TOT / 16) * (NST / 16);
    wmma_gemm_xwt<<<waves * 32 / 256, 256, 0, stream>>>(
        x, B_w, nullptr, Bm, BL_TOT, NST, DIN, 0);
  }
  // Cm = x @ C_w^T
  {
    int waves = (BL_TOT / 16) * (NST / 16);
    wmma_gemm_xwt<<<waves * 32 / 256, 256, 0, stream>>>(
        x, C_w, nullptr, Cm, BL_TOT, NST, DIN, 0);
  }
  // G1 = gelu(x @ dt_w1^T + b1)  (M=1024, N=256, K=1024)
  {
    int waves = (BL_TOT / 16) * (HMID / 16);
    wmma_gemm_xwt<<<waves * 32 / 256, 256, 0, stream>>>(
        x, dt_w1, dt_b1, G1, BL_TOT, HMID, DIN, 1);
  }
  // dt = softplus(G1 @ dt_w2^T + b2) * 0.1  (M=1024, N=1024, K=256)
  {
    int waves = (BL_TOT / 16) * (DIN / 16);
    wmma_gemm_xwt<<<waves * 32 / 256, 256, 0, stream>>>(
        G1, dt_w2, dt_b2, dt, BL_TOT, DIN, HMID, 2);
  }

  // Fused streaming pass: one wave per (bl,d) row -> BL*DIN waves.
  {
    const long long rows   = (long long)BL_TOT * DIN;       // 1,048,576 waves
    const long long blocks = rows * 32 / 256;               // 131,072 blocks
    fused_state_kernel<<<(int)blocks, 256, 0, stream>>>(
        x, h, A, Bm, Cm, Dv, dt, y, hn);
  }
}